// EquivariantLayer_8177617731956
// MI455X (gfx1250) — compile-verified
//
#include <hip/hip_runtime.h>

typedef float v2f __attribute__((ext_vector_type(2)));
typedef float v8f __attribute__((ext_vector_type(8)));

#define PI2F 6.28318530717958647692f

// ---------------------------------------------------------------------------
// gfx1250 async helpers
// ---------------------------------------------------------------------------
__device__ __forceinline__ void async_b128_to_lds(uint32_t lds_off, uint64_t gaddr) {
    // GLOBAL_LOAD_ASYNC_TO_LDS_B128: VDST = LDS byte offset (VGPR),
    // VADDR = 64-bit global address.  Tracked with ASYNCcnt.
    asm volatile("global_load_async_to_lds_b128 %0, %1, off"
                 :: "v"(lds_off), "v"(gaddr)
                 : "memory");
}

__device__ __forceinline__ void wait_async0() {
#if __has_builtin(__builtin_amdgcn_s_wait_asynccnt)
    __builtin_amdgcn_s_wait_asynccnt(0);
#else
    asm volatile("s_wait_asynccnt 0x0" ::: "memory");
#endif
}

// ---------------------------------------------------------------------------
// Basis matrix construction (deterministic, rebuilt every launch)
// ---------------------------------------------------------------------------

// WC[256][128]: column rfft (N=256) pack: cols 0..63 = cos, 64..127 = -sin
__global__ void init_wc(float* __restrict__ W) {
    int idx = blockIdx.x * blockDim.x + threadIdx.x;   // 32768
    int y = idx >> 7, n = idx & 127, v = n & 63;
    int ph = (y * v) & 255;
    float th = PI2F * (float)ph / 256.0f;
    W[idx] = (n < 64) ? __cosf(th) : -__sinf(th);
}

// WR[256][512]: real-embedded forward row DFT (N=256, 128 truncated freqs),
// scaled by (N2/N1)^2 = 0.25, row u=64 zeroed (Nyquist kill).
__global__ void init_wr(float* __restrict__ W) {
    int idx = blockIdx.x * blockDim.x + threadIdx.x;   // 131072
    int r = idx >> 9, k = idx & 511;
    int rr = r & 127;
    int ku = (rr > 64) ? rr - 128 : rr;                // signed frequency
    int x = k & 255;
    int ph = (ku * x) % 256; if (ph < 0) ph += 256;
    float th = PI2F * (float)ph / 256.0f;
    float c = 0.25f * __cosf(th), s = 0.25f * __sinf(th);
    bool re_row = (r < 128), k_lo = (k < 256);
    // F_re = H_re*cos + H_im*sin ; F_im = -H_re*sin + H_im*cos
    float val = re_row ? (k_lo ? c : s) : (k_lo ? -s : c);
    W[idx] = (rr == 64) ? 0.0f : val;
}

// WR128[256][256]: real-embedded forward row DFT (N=128), unscaled.
__global__ void init_wr128(float* __restrict__ W) {
    int idx = blockIdx.x * blockDim.x + threadIdx.x;   // 65536
    int r = idx >> 8, k = idx & 255;
    int u = r & 127, x = k & 127;
    int ph = (u * x) & 127;
    float th = PI2F * (float)ph / 128.0f;
    float c = __cosf(th), s = __sinf(th);
    bool re_row = (r < 128), k_lo = (k < 128);
    W[idx] = re_row ? (k_lo ? c : s) : (k_lo ? -s : c);
}

// VR[256][256]: real-embedded inverse row DFT (N=128), 1/128 normalized.
__global__ void init_vr(float* __restrict__ W) {
    int idx = blockIdx.x * blockDim.x + threadIdx.x;   // 65536
    int r = idx >> 8, k = idx & 255;
    int x = r & 127, u = k & 127;
    int ph = (u * x) & 127;
    float th = PI2F * (float)ph / 128.0f;
    float c = __cosf(th) * (1.0f / 128.0f), s = __sinf(th) * (1.0f / 128.0f);
    bool re_row = (r < 128), k_lo = (k < 128);
    // T_re = F_re*cos - F_im*sin ; T_im = F_re*sin + F_im*cos
    W[idx] = re_row ? (k_lo ? c : -s) : (k_lo ? s : c);
}

// BC[128][128]: Hermitian irfft combine weights (last axis, N=128).
__global__ void init_bc(float* __restrict__ W) {
    int idx = blockIdx.x * blockDim.x + threadIdx.x;   // 16384
    int k = idx >> 7, y = idx & 127, v = k & 63;
    float sc = ((v == 0) ? 1.0f : 2.0f) * (1.0f / 128.0f);
    int ph = (v * y) & 127;
    float th = PI2F * (float)ph / 128.0f;
    W[idx] = (k < 64) ? sc * __cosf(th) : -sc * __sinf(th);
}

// WC128[128][128]: column rfft (N=128) pack
__global__ void init_wc128(float* __restrict__ W) {
    int idx = blockIdx.x * blockDim.x + threadIdx.x;   // 16384
    int y = idx >> 7, n = idx & 127, v = n & 63;
    int ph = (y * v) & 127;
    float th = PI2F * (float)ph / 128.0f;
    W[idx] = (n < 64) ? __cosf(th) : -__sinf(th);
}

// ---------------------------------------------------------------------------
// D4 symmetrization of the 128x128 kernel taps (8 periodic symmetries)
// ---------------------------------------------------------------------------
__global__ void symmetrize(const float* __restrict__ kin, float* __restrict__ sym) {
    int idx = blockIdx.x * blockDim.x + threadIdx.x;   // 16384 per image
    int j = blockIdx.y;                                // 0..511  (c*32+d)
    const float* t = kin + (size_t)j * 16384;
    int a = idx >> 7, b = idx & 127;
    int na = (128 - a) & 127, nb = (128 - b) & 127;
    float s = t[a * 128 + b]   + t[nb * 128 + a]  + t[na * 128 + nb] + t[b * 128 + na]
            + t[b * 128 + a]   + t[a * 128 + nb]  + t[nb * 128 + na] + t[na * 128 + b];
    sym[(size_t)j * 16384 + idx] = s * 0.125f;
}

// ---------------------------------------------------------------------------
// Layout repacks between GEMM stages
// ---------------------------------------------------------------------------
__global__ void repack_stack(const float* __restrict__ in, float* __restrict__ out,
                             int X, int imgElems) {
    int idx = blockIdx.x * blockDim.x + threadIdx.x;   // over imgElems = 2*X*64
    int img = blockIdx.y;
    int k = idx >> 6, n = idx & 63;
    int x   = (k < X) ? k : (k - X);
    int off = (k < X) ? 0 : 64;
    out[(size_t)img * imgElems + idx] =
        in[(size_t)img * imgElems + (size_t)x * 128 + off + n];
}

__global__ void repack_cat(const float* __restrict__ in, float* __restrict__ out) {
    int idx = blockIdx.x * blockDim.x + threadIdx.x;   // over 16384
    int img = blockIdx.y;
    int x = idx >> 7, n = idx & 127;
    size_t src = (n < 64) ? ((size_t)x * 64 + n)
                          : ((size_t)(128 + x) * 64 + (n - 64));
    out[(size_t)img * 16384 + idx] = in[(size_t)img * 16384 + src];
}

// ---------------------------------------------------------------------------
// Batched row-major f32 GEMM with async-to-LDS B-panel staging.
// Block = 128 threads (4 waves). blockIdx.x -> 16-col N tile,
// blockIdx.y -> 64-row M supertile (one 16-row tile per wave), blockIdx.z -> img.
// The K x 16 B panel is staged into LDS once per workgroup with
// GLOBAL_LOAD_ASYNC_TO_LDS_B128 (ASYNCcnt), then all 4 waves feed WMMA from LDS.
// Requires: M % 64 == 0, N % 16 == 0, K % 32 == 0, K <= 512, B 16B-aligned.
// ---------------------------------------------------------------------------
__global__ void wmma_gemm_f32(const float* __restrict__ A, int sA, int lda,
                              const float* __restrict__ B, int sB, int ldb,
                              float* __restrict__ C, int sC, int ldc, int K) {
    __shared__ __align__(16) float sBp[512 * 16];      // up to 32 KB

    const int tid   = threadIdx.x;                     // 0..127
    const int wave  = tid >> 5;
    const int lane  = tid & 31;
    const int l15   = lane & 15;
    const int khalf = (lane >> 4) << 1;
    const int tn  = blockIdx.x << 4;
    const int tm  = (blockIdx.y << 6) + (wave << 4);
    const int img = blockIdx.z;

    // ---- async-stage B panel [K][16] -> LDS (b128 = 4 floats per copy) ----
    const float* Bg = B + (size_t)img * sB + tn;
    const int groups = K << 2;                         // K*4 16-byte groups (mult of 128)
    for (int g = tid; g < groups; g += 128) {
        const int row = g >> 2;
        const int c4  = (g & 3) << 2;
        uint64_t gaddr = (uint64_t)(uintptr_t)(Bg + (size_t)row * ldb + c4);
        uint32_t laddr = (uint32_t)(uintptr_t)&sBp[(row << 4) + c4];
        async_b128_to_lds(laddr, gaddr);
    }
    wait_async0();
    __syncthreads();

    // ---- WMMA main loop: A from global (prefetched), B from LDS ----
    const float* Ar = A + (size_t)img * sA + (size_t)(tm + l15) * lda;
    v8f acc = {};
#pragma unroll 4
    for (int k = 0; k < K; k += 4) {
        v2f a, b;
        a.x = Ar[k + khalf];
        a.y = Ar[k + khalf + 1];
        b.x = sBp[((k + khalf) << 4) + l15];
        b.y = sBp[((k + khalf + 1) << 4) + l15];
        __builtin_prefetch(Ar + k + 32, 0, 0);         // global_prefetch_b8
        acc = __builtin_amdgcn_wmma_f32_16x16x4_f32(false, a, false, b,
                                                    (short)0, acc, false, false);
    }
    float* Cr = C + (size_t)img * sC + (size_t)(tm + ((lane >> 4) << 3)) * ldc + (tn + l15);
#pragma unroll
    for (int r = 0; r < 8; ++r) Cr[(size_t)r * ldc] = acc[r];
}

// ---------------------------------------------------------------------------
// Per-frequency complex einsum  G[b,d] = sum_c F[b,c] * K[c,d]
// A = [F_re(8x16); F_im(8x16)] (16x16); P = A*K_re, Q = A*K_im (each 16x32 as
// two 16x16 WMMA tiles, K=16 in 4 steps). Recombine through LDS:
//   G_re[b] = P[b] - Q[b+8] ;  G_im[b] = Q[b] + P[b+8]
// ---------------------------------------------------------------------------
__global__ void spectral_einsum(const float* __restrict__ F,   // 128 x (256*64)
                                const float* __restrict__ Kk,  // 512 x (256*64)
                                float* __restrict__ G) {       // 256 x (256*64)
    __shared__ float ldsP[8][2][256];
    __shared__ float ldsQ[8][2][256];
    const int tid  = threadIdx.x;
    const int wave = tid >> 5;
    const int lane = tid & 31;
    const int l15  = lane & 15;
    const int hi   = lane >> 4;
    const int khalf = hi << 1;
    const int fq = (blockIdx.x << 3) + wave;          // 0..8191
    const int u = fq >> 6, v = fq & 63;
    const int fo = u * 64 + v;                        // offset within re/im plane

    const int m    = l15;                             // A row this lane feeds
    const int bidx = m & 7;                           // batch index
    const int imo  = (m < 8) ? 0 : 8192;              // re / im plane

    v8f p0 = {}, p1 = {}, q0 = {}, q1 = {};
#pragma unroll
    for (int k0 = 0; k0 < 16; k0 += 4) {
        const int c0 = k0 + khalf;                    // channel (K dim)
        v2f a;
        a.x = F[(size_t)(bidx * 16 + c0)     * 16384 + imo + fo];
        a.y = F[(size_t)(bidx * 16 + c0 + 1) * 16384 + imo + fo];
        const int d0 = l15, d1 = 16 + l15;
        size_t ka0 = (size_t)(c0 * 32 + d0)       * 16384 + fo;
        size_t kb0 = (size_t)((c0 + 1) * 32 + d0) * 16384 + fo;
        size_t ka1 = (size_t)(c0 * 32 + d1)       * 16384 + fo;
        size_t kb1 = (size_t)((c0 + 1) * 32 + d1) * 16384 + fo;
        v2f br0, bi0, br1, bi1;
        br0.x = Kk[ka0];        br0.y = Kk[kb0];
        bi0.x = Kk[ka0 + 8192]; bi0.y = Kk[kb0 + 8192];
        br1.x = Kk[ka1];        br1.y = Kk[kb1];
        bi1.x = Kk[ka1 + 8192]; bi1.y = Kk[kb1 + 8192];
        p0 = __builtin_amdgcn_wmma_f32_16x16x4_f32(false, a, false, br0, (short)0, p0, false, false);
        q0 = __builtin_amdgcn_wmma_f32_16x16x4_f32(false, a, false, bi0, (short)0, q0, false, false);
        p1 = __builtin_amdgcn_wmma_f32_16x16x4_f32(false, a, false, br1, (short)0, p1, false, false);
        q1 = __builtin_amdgcn_wmma_f32_16x16x4_f32(false, a, false, bi1, (short)0, q1, false, false);
    }
    // spill C tiles to LDS (row = r + hi*8 per documented C/D layout)
#pragma unroll
    for (int r = 0; r < 8; ++r) {
        const int row = r + (hi << 3);
        ldsP[wave][0][row * 16 + l15] = p0[r];
        ldsP[wave][1][row * 16 + l15] = p1[r];
        ldsQ[wave][0][row * 16 + l15] = q0[r];
        ldsQ[wave][1][row * 16 + l15] = q1[r];
    }
    __syncthreads();
    const int t = hi;                 // N-tile handled by this half-wave
    const int col = l15;
#pragma unroll
    for (int b = 0; b < 8; ++b) {
        float pre = ldsP[wave][t][b * 16 + col];
        float pim = ldsP[wave][t][(b + 8) * 16 + col];
        float qre = ldsQ[wave][t][b * 16 + col];
        float qim = ldsQ[wave][t][(b + 8) * 16 + col];
        float gre = pre - qim;
        float gim = qre + pim;
        const int d = (t << 4) + col;
        const int o = b * 32 + d;     // output image (b,d)
        G[(size_t)o * 16384 + fo]        = gre;
        G[(size_t)o * 16384 + 8192 + fo] = gim;
    }
}

// ---------------------------------------------------------------------------
// Launch: full pipeline  (all fp32, all GEMMs via WMMA + async-LDS staging)
// ---------------------------------------------------------------------------
extern "C" void kernel_launch(void* const* d_in, const int* in_sizes, int n_in,
                              void* d_out, int out_size, void* d_ws, size_t ws_size,
                              hipStream_t stream) {
    (void)in_sizes; (void)n_in; (void)out_size; (void)ws_size;
    const float* f   = (const float*)d_in[0];   // (8,16,256,256)
    const float* ker = (const float*)d_in[1];   // (1,16,32,128,128)
    float* out = (float*)d_out;                 // out1 (8,16,128,128) ++ out2 (8,32,128,128)
    float* ws  = (float*)d_ws;

    // workspace layout (floats); peak ~102 MB
    float* WC    = ws;                 // 256x128
    float* WR    = ws + 32768;         // 256x512
    float* VR    = ws + 163840;        // 256x256
    float* BC    = ws + 229376;        // 128x128
    float* WC128 = ws + 245760;        // 128x128
    float* WR128 = ws + 262144;        // 256x256
    float* RA    = ws + 327680;        // 8388608-float region
    float* RB    = RA + 8388608;       // 8388608-float region
    float* RC    = RB + 8388608;       // 8388608-float region

    // basis matrices
    init_wc   <<<128, 256, 0, stream>>>(WC);
    init_wr   <<<512, 256, 0, stream>>>(WR);
    init_vr   <<<256, 256, 0, stream>>>(VR);
    init_bc   <<< 64, 256, 0, stream>>>(BC);
    init_wc128<<< 64, 256, 0, stream>>>(WC128);
    init_wr128<<<256, 256, 0, stream>>>(WR128);

    // ---- kernel spectrum: Ks = rfft2(symmetrize(kernel)), 512 images ----
    symmetrize<<<dim3(64, 512), 256, 0, stream>>>(ker, RA);                    // sym -> RA
    wmma_gemm_f32<<<dim3(8, 2, 512), 128, 0, stream>>>(RA, 16384, 128,
                                                       WC128, 0, 128,
                                                       RB, 16384, 128, 128);   // Hkp -> RB
    repack_stack<<<dim3(64, 512), 256, 0, stream>>>(RB, RA, 128, 16384);       // Hks -> RA
    wmma_gemm_f32<<<dim3(4, 4, 512), 128, 0, stream>>>(WR128, 0, 256,
                                                       RA, 16384, 64,
                                                       RB, 16384, 64, 256);    // Ks  -> RB

    // ---- input spectrum: Fs = trunc(rfft2(f)) * 0.25, 128 images ----
    wmma_gemm_f32<<<dim3(8, 4, 128), 128, 0, stream>>>(f, 65536, 256,
                                                       WC, 0, 128,
                                                       RC, 32768, 128, 256);   // Hp  -> RC[0..]
    repack_stack<<<dim3(128, 128), 256, 0, stream>>>(RC, RC + 4194304, 256, 32768); // Hs
    wmma_gemm_f32<<<dim3(4, 4, 128), 128, 0, stream>>>(WR, 0, 512,
                                                       RC + 4194304, 32768, 64,
                                                       RC, 16384, 64, 512);    // Fs  -> RC[0..]

    // ---- per-frequency complex einsum: Gs (256 images) ----
    spectral_einsum<<<1024, 256, 0, stream>>>(RC, RB, RA);                     // Gs -> RA[0..]

    // ---- out1 = irfft2(Fs) ----
    wmma_gemm_f32<<<dim3(4, 4, 128), 128, 0, stream>>>(VR, 0, 256,
                                                       RC, 16384, 64,
                                                       RC + 2097152, 16384, 64, 256); // Ts
    repack_cat<<<dim3(64, 128), 256, 0, stream>>>(RC + 2097152, RC + 4194304); // Tcat
    wmma_gemm_f32<<<dim3(8, 2, 128), 128, 0, stream>>>(RC + 4194304, 16384, 128,
                                                       BC, 0, 128,
                                                       out, 16384, 128, 128);  // out1

    // ---- out2 = irfft2(Gs) ----
    wmma_gemm_f32<<<dim3(4, 4, 256), 128, 0, stream>>>(VR, 0, 256,
                                                       RA, 16384, 64,
                                                       RA + 4194304, 16384, 64, 256); // Tg
    repack_cat<<<dim3(64, 256), 256, 0, stream>>>(RA + 4194304, RC);           // Tgcat
    wmma_gemm_f32<<<dim3(8, 2, 256), 128, 0, stream>>>(RC, 16384, 128,
                                                       BC, 0, 128,
                                                       out + 2097152, 16384, 128, 128); // out2
}